// JaggedAppend_78005196030024
// MI455X (gfx1250) — compile-verified
//
#include <hip/hip_runtime.h>

// Jagged append: out = concat_i( values[seg_i] ++ suffix_mat[i] )
//
// Pure data movement (0 FLOPs). MI455X roofline: ~150 MB total traffic at
// 23.3 TB/s -> ~6.5 us floor. One 256-thread block (8 wave32) per segment;
// contiguous b128 copies with non-temporal hints (each byte touched once, so
// don't pollute the 192 MB L2).
//
// new_start - prev_old = seg*256 (multiple of 4 elements) => src and dst
// share alignment mod 16B; after a <=3 element scalar head the whole body is
// 16B global_load_b128/global_store_b128.
//
// All intra-segment indices are 32-bit (seg_len <= 3072) so addressing uses
// scalar-base + 32-bit lane offset; body unrolled x4 for 4 outstanding b128
// loads per wave.

typedef __attribute__((ext_vector_type(4))) float v4f;

#define SUF 256
#define BLOCK 256

__global__ __launch_bounds__(BLOCK) void jagged_append_kernel(
    const float* __restrict__ values,
    const long long* __restrict__ prefix_sum,   // inclusive prefix sum, int64
    const float* __restrict__ suffix_mat,       // [B, SUF]
    float* __restrict__ out)
{
    const int seg = blockIdx.x;

    // Wave-uniform segment bounds -> scalar loads.
    const long long cur  = prefix_sum[seg];
    const long long prev = (seg > 0) ? prefix_sum[seg - 1] : 0ll;
    const int       seg_len   = (int)(cur - prev);            // <= 3072
    const long long new_start = prev + (long long)seg * SUF;

    const float* __restrict__ src = values + prev;
    float* __restrict__ dst       = out + new_start;

    const int tid = threadIdx.x;

    // ---- head: bring src/dst (same alignment mod 16) up to 16B alignment ----
    const int mis  = (int)(prev & 3);
    int head = mis ? (4 - mis) : 0;
    if (head > seg_len) head = seg_len;
    if (tid < head) {
        dst[tid] = src[tid];
    }

    // ---- body: 16B vector copy, non-temporal, unrolled x4 for MLP ----
    const int n4 = (seg_len - head) >> 2;
    const v4f* __restrict__ src4 = (const v4f*)(src + head);
    v4f* __restrict__ dst4       = (v4f*)(dst + head);

    int i = tid;
    for (; i + 3 * BLOCK < n4; i += 4 * BLOCK) {
        v4f a = __builtin_nontemporal_load(&src4[i]);
        v4f b = __builtin_nontemporal_load(&src4[i + BLOCK]);
        v4f c = __builtin_nontemporal_load(&src4[i + 2 * BLOCK]);
        v4f d = __builtin_nontemporal_load(&src4[i + 3 * BLOCK]);
        __builtin_nontemporal_store(a, &dst4[i]);
        __builtin_nontemporal_store(b, &dst4[i + BLOCK]);
        __builtin_nontemporal_store(c, &dst4[i + 2 * BLOCK]);
        __builtin_nontemporal_store(d, &dst4[i + 3 * BLOCK]);
    }
    for (; i < n4; i += BLOCK) {
        v4f v = __builtin_nontemporal_load(&src4[i]);
        __builtin_nontemporal_store(v, &dst4[i]);
    }

    // ---- tail: remaining 0..3 scalars ----
    const int done = head + (n4 << 2);
    const int rem  = seg_len - done;               // 0..3
    if (tid < rem) {
        dst[done + tid] = src[done + tid];
    }

    // ---- suffix: 256 floats, one per thread, coalesced ----
    const float* __restrict__ srow = suffix_mat + (long long)seg * SUF;
    float* __restrict__ dsuf       = dst + seg_len;
    {
        float v = __builtin_nontemporal_load(&srow[tid]);
        __builtin_nontemporal_store(v, &dsuf[tid]);
    }
}

extern "C" void kernel_launch(void* const* d_in, const int* in_sizes, int n_in,
                              void* d_out, int out_size, void* d_ws, size_t ws_size,
                              hipStream_t stream) {
    const float* values          = (const float*)d_in[0];
    const long long* prefix_sum  = (const long long*)d_in[1];   // int64 per reference
    const float* suffix_mat      = (const float*)d_in[2];
    float* out                   = (float*)d_out;

    const int B = in_sizes[1];   // number of segments (8192)

    jagged_append_kernel<<<dim3(B), dim3(BLOCK), 0, stream>>>(
        values, prefix_sum, suffix_mat, out);
}